// DeltaEncoder_34969623724327
// MI455X (gfx1250) — compile-verified
//
#include <hip/hip_runtime.h>
#include <hip/hip_bf16.h>

typedef __attribute__((ext_vector_type(2))) float v2f;
typedef __attribute__((ext_vector_type(8))) float v8f;

#define BB 32
#define LL 4096
#define CC 8
#define TT 64
#define NPOS (BB * CC * LL)  // 1048576

// ---------------- pass 0: zero the f64 accumulators ----------------
__global__ void de_init_ws(double* ws) {
    if (threadIdx.x == 0) { ws[0] = 0.0; ws[1] = 0.0; }
}

// ---------------- pass 1: global sum / sumsq of delta --------------
__global__ void __launch_bounds__(256)
de_reduce(const float* __restrict__ x, double* __restrict__ ws) {
    __shared__ double ssum[256];
    __shared__ double ssq[256];
    double s = 0.0, q = 0.0;
    int stride = gridDim.x * blockDim.x;
    for (int p = blockIdx.x * blockDim.x + threadIdx.x; p < NPOS; p += stride) {
        int b = p >> 15;           // / (C*L)
        int c = (p >> 12) & 7;     // / L % C
        int l = p & 4095;          // % L
        int idx = ((b << 12) + l) * 8 + c;   // [B,L,C] flat
        float cur = x[idx];
        float dlt = (l == 0) ? 0.0f : (cur - x[idx - 8]);
        s += (double)dlt;
        q += (double)dlt * (double)dlt;
    }
    ssum[threadIdx.x] = s;
    ssq[threadIdx.x]  = q;
    __syncthreads();
    for (int off = 128; off > 0; off >>= 1) {
        if ((int)threadIdx.x < off) {
            ssum[threadIdx.x] += ssum[threadIdx.x + off];
            ssq[threadIdx.x]  += ssq[threadIdx.x + off];
        }
        __syncthreads();
    }
    if (threadIdx.x == 0) {
        atomicAdd(&ws[0], ssum[0]);   // global_atomic_add_f64
        atomicAdd(&ws[1], ssq[0]);
    }
}

// ---------------- pass 2: fused BN + encoder(WMMA) + LIF -----------
// One wave handles 16 consecutive l (fixed b,c) x all 64 timesteps.
// enc tile (16t x 16pos) via V_WMMA_F32_16X16X4_F32:
//   A[m,k] = {w[t0+m], bias[t0+m], 0, 0}   (M=t)
//   B[k,n] = {d[n],    1,          0, 0}   (N=pos)
//   D[m,n] = w[t0+m]*d[n] + bias[t0+m] = enc
__global__ void __launch_bounds__(256)
de_main(const float* __restrict__ x,
        const float* __restrict__ gamma,
        const float* __restrict__ beta,
        const float* __restrict__ ew,
        const float* __restrict__ ebias,
        const double* __restrict__ ws,
        float* __restrict__ out) {
    // BN affine folded into scale/shift
    const double invN = 1.0 / (double)NPOS;
    double mean_d = ws[0] * invN;
    double var_d  = ws[1] * invN - mean_d * mean_d;
    float rs    = (float)(1.0 / sqrt(var_d + 1e-5));
    float scale = gamma[0] * rs;
    float shift = beta[0] - (float)mean_d * scale;

    int lane = threadIdx.x & 31;
    int wave = threadIdx.x >> 5;
    int tile = blockIdx.x * 8 + wave;        // 65536 tiles total
    int b   = tile >> 11;                    // / (C * L/16)
    int c   = (tile >> 8) & 7;
    int l16 = (tile & 255) << 4;
    int n   = lane & 15;                     // position within tile
    int l   = l16 + n;
    bool lo = lane < 16;

    // delta + BN for this lane's position (high lanes duplicate; harmless)
    int idx = ((b << 12) + l) * 8 + c;
    float cur = x[idx];
    float dlt = (l == 0) ? 0.0f : (cur - x[idx - 8]);
    float d = dlt * scale + shift;

    // B operand: row K=0 = d, K=1 = ones, K=2/3 = 0 (high lanes)
    v2f bop;
    bop.x = lo ? d : 0.0f;
    bop.y = lo ? 1.0f : 0.0f;

    const size_t tstride = (size_t)CC * LL;  // 32768
    size_t ob = (((size_t)b * TT) * CC + c) * (size_t)LL + (size_t)l;

    float vcar = 0.0f;                       // membrane carry, replicated per half
    #pragma unroll
    for (int tt = 0; tt < 4; ++tt) {
        int t0 = tt * 16;
        int ti = t0 + n;                     // in-bounds for all lanes
        v2f aop;
        aop.x = lo ? ew[ti]    : 0.0f;       // A row m: K=0 -> w[t0+m]
        aop.y = lo ? ebias[ti] : 0.0f;       // K=1 -> bias[t0+m]

        v8f acc = {};
        acc = __builtin_amdgcn_wmma_f32_16x16x4_f32(
            /*neg_a=*/false, aop, /*neg_b=*/false, bop,
            /*c_mod=*/(short)0, acc, /*reuse_a=*/false, /*reuse_b=*/false);
        // acc[r]: lanes 0-15 -> enc[t0+r][n]; lanes 16-31 -> enc[t0+8+r][n]

        float s1[8], s2[8];
        // phase 1: t0..t0+7 (valid in lanes 0-15)
        float v = vcar;
        #pragma unroll
        for (int r = 0; r < 8; ++r) {
            float u = (v + acc[r]) * 0.5f;   // v += (enc - v)/TAU, TAU=2
            bool sp = (u >= 1.0f);           // v - V_TH >= 0
            s1[r] = sp ? 1.0f : 0.0f;
            v = sp ? 0.0f : u;               // hard reset
        }
        // pass v(t0+7) from low half to all lanes
        float vmid = __shfl(v, n, 32);
        // phase 2: t0+8..t0+15 (valid in lanes 16-31)
        float v2 = vmid;
        #pragma unroll
        for (int r = 0; r < 8; ++r) {
            float u = (v2 + acc[r]) * 0.5f;
            bool sp = (u >= 1.0f);
            s2[r] = sp ? 1.0f : 0.0f;
            v2 = sp ? 0.0f : u;
        }
        // carry v(t0+15) from high half back to all lanes
        vcar = __shfl(v2, n + 16, 32);

        int tb = t0 + (lo ? 0 : 8);
        #pragma unroll
        for (int r = 0; r < 8; ++r) {
            out[ob + (size_t)(tb + r) * tstride] = lo ? s1[r] : s2[r];
        }
    }
}

extern "C" void kernel_launch(void* const* d_in, const int* in_sizes, int n_in,
                              void* d_out, int out_size, void* d_ws, size_t ws_size,
                              hipStream_t stream) {
    const float* x     = (const float*)d_in[0];   // [32,4096,8]
    const float* gamma = (const float*)d_in[1];   // [1]
    const float* beta  = (const float*)d_in[2];   // [1]
    const float* ew    = (const float*)d_in[3];   // [64,1]
    const float* eb    = (const float*)d_in[4];   // [64]
    float* out = (float*)d_out;                   // [32,64,8,4096]
    double* ws = (double*)d_ws;                   // 2 x f64 accumulators

    de_init_ws<<<1, 32, 0, stream>>>(ws);
    de_reduce<<<512, 256, 0, stream>>>(x, ws);
    // 65536 waves = 8192 blocks x 8 waves; exact tiling, no divergence
    de_main<<<8192, 256, 0, stream>>>(x, gamma, beta, ew, eb, ws, out);
}